// CapsLayer_70489003262278
// MI455X (gfx1250) — compile-verified
//
#include <hip/hip_runtime.h>

#define C_CAPS 16
#define N_ROUTE 1152
#define IN_CH 64
#define OUT_CH 64
#define BATCH 64
#define NB 8                      // n-tiles per GEMM workgroup
#define NCHUNK (N_ROUTE / NB)     // 144
#define XS_STRIDE 68              // == 4 mod 64 -> conflict-free A b64 loads
#define WF_STRIDE 160             // == 32 mod 64 -> lanes 0-15 / 16-31 on disjoint bank halves

typedef __attribute__((ext_vector_type(2))) float v2f;
typedef __attribute__((ext_vector_type(8))) float v8f;

// ---------------------------------------------------------------------------
// Kernel 1: priors[c][b][n][o] = sum_i x[b][n][i] * w[c][n][i][o]
// Block = (n-chunk of 8, c); per n a 64x64x64 f32 GEMM on
// V_WMMA_F32_16X16X4_F32 (4 waves x 4 o-tiles x 16 K-steps).
// Also accumulates sum_n priors in registers -> partials (fused routing init).
// LDS layouts are WMMA-fragment native:
//   xs[b][i]    stride 68: A-frag  = b64 at  bb*68 + 4*kq + 2*lhi   (all 64 banks hit once)
//   wfrag rows (2*kq+h) stride 160: B-frag = b64 at row*160 + 2*oo  (bank halves disjoint)
// ---------------------------------------------------------------------------
__global__ __launch_bounds__(128) void priors_gemm_kernel(
    const float* __restrict__ x, const float* __restrict__ w,
    float* __restrict__ priors, float* __restrict__ partials) {
  const int chunk = blockIdx.x;  // 0..143
  const int c = blockIdx.y;
  const int n0 = chunk * NB;
  __shared__ float xs[BATCH * XS_STRIDE];   // 17 KB
  __shared__ float wfrag[32 * WF_STRIDE];   // 20 KB
  const int t = threadIdx.x;
  const int wave = t >> 5;
  const int lane = t & 31;
  const int lrow = lane & 15;
  const int lhi = lane >> 4;

  v8f psum[4] = {};

  for (int j = 0; j < NB; ++j) {
    const int n = n0 + j;
    // ---- stage x tile: x[b][n][0..63] -> xs[b][i] (padded rows) ----
    for (int idx = t; idx < (BATCH * IN_CH) / 4; idx += 128) {
      const int b = idx >> 4;
      const int i4 = idx & 15;
      const float4 val =
          ((const float4*)(x + ((size_t)b * N_ROUTE + n) * IN_CH))[i4];
      *(float4*)(xs + b * XS_STRIDE + i4 * 4) = val;
    }
    // ---- stage W tile into fragment layout:
    // W[i][o] -> wfrag[(i/4)*2 + (i%4)/2][o*2 + (i%2)] ----
    const float* wg = w + ((size_t)c * N_ROUTE + n) * (IN_CH * OUT_CH);
    for (int idx = t; idx < (IN_CH * OUT_CH) / 4; idx += 128) {
      const int i = idx >> 4;
      const int o4 = idx & 15;
      const float4 val = ((const float4*)wg)[idx];
      const int row = (i >> 2) * 2 + ((i >> 1) & 1);
      float* dst = wfrag + row * WF_STRIDE + (o4 << 3) + (i & 1);
      dst[0] = val.x;
      dst[2] = val.y;
      dst[4] = val.z;
      dst[6] = val.w;
    }
    // ---- prefetch next n-tile while this one computes ----
    if (j + 1 < NB) {
      __builtin_prefetch(wg + IN_CH * OUT_CH + t * 32, 0, 3);
      if (t < BATCH)
        __builtin_prefetch(x + ((size_t)t * N_ROUTE + n + 1) * IN_CH, 0, 3);
    }
    __syncthreads();

    v8f acc[4] = {};
    const float* arow = xs + ((wave << 4) + lrow) * XS_STRIDE;
#pragma unroll
    for (int kq = 0; kq < 16; ++kq) {
      // A frag (16x4): lane L, vgpr v -> M = L&15, K = 4kq + 2*(L>>4) + v
      const v2f a = *(const v2f*)(arow + (kq << 2) + 2 * lhi);
      const float* brow = wfrag + ((kq << 1) + lhi) * WF_STRIDE;
#pragma unroll
      for (int ot = 0; ot < 4; ++ot) {
        // B frag (4x16): lane L, vgpr v -> K = 4kq + 2*(L>>4) + v, N = L&15
        const v2f bf = *(const v2f*)(brow + (((ot << 4) + lrow) << 1));
        acc[ot] = __builtin_amdgcn_wmma_f32_16x16x4_f32(
            false, a, false, bf, (short)0, acc[ot], false, false);
      }
    }

    // ---- store priors tile; fold into running n-sum ----
    float* pg = priors + ((size_t)c * BATCH * N_ROUTE + n) * OUT_CH;
#pragma unroll
    for (int ot = 0; ot < 4; ++ot) {
      const int oo = (ot << 4) + lrow;
#pragma unroll
      for (int r = 0; r < 8; ++r) {
        const int bb = (wave << 4) + (lhi << 3) + r;  // C layout: M = r + 8*(L>>4)
        pg[(size_t)bb * N_ROUTE * OUT_CH + oo] = acc[ot][r];
      }
      psum[ot] += acc[ot];
    }
    __syncthreads();  // LDS reuse guard for next n
  }

  // partials[c][chunk][b][o] = sum over this chunk's 8 n of priors
  float* pp = partials + (size_t)(c * NCHUNK + chunk) * (BATCH * OUT_CH);
#pragma unroll
  for (int ot = 0; ot < 4; ++ot) {
    const int oo = (ot << 4) + lrow;
#pragma unroll
    for (int r = 0; r < 8; ++r) {
      const int bb = (wave << 4) + (lhi << 3) + r;
      pp[bb * OUT_CH + oo] = psum[ot][r];
    }
  }
}

// ---------------------------------------------------------------------------
// Kernel 2: v0 = squash(mean_n priors) from the 144-chunk partials (38 MB
// instead of re-reading 302 MB of priors). Also zeroes logits.
// ---------------------------------------------------------------------------
__global__ __launch_bounds__(256) void routing_init_kernel(
    const float* __restrict__ partials, float* __restrict__ vbuf,
    float* __restrict__ logits) {
  const int b = blockIdx.x;
  const int c = blockIdx.y;
  const int t = threadIdx.x;
  const int o4 = (t & 15) << 2;
  const int grp = t >> 4;  // 0..15

  float4 a4 = {0.f, 0.f, 0.f, 0.f};
  for (int ch = grp; ch < NCHUNK; ch += 16) {
    const float4 p = *(const float4*)(
        partials + ((size_t)(c * NCHUNK + ch) * BATCH + b) * OUT_CH + o4);
    a4.x += p.x; a4.y += p.y; a4.z += p.z; a4.w += p.w;
  }
  __shared__ float part[16][OUT_CH];
  *(float4*)&part[grp][o4] = a4;

  float* lg = logits + (size_t)(c * BATCH + b) * N_ROUTE;
  for (int nn = t; nn < N_ROUTE; nn += 256) lg[nn] = 0.0f;
  __syncthreads();

  __shared__ float s_sh[OUT_CH];
  __shared__ float sq[OUT_CH];
  if (t < OUT_CH) {
    float s = 0.f;
#pragma unroll
    for (int g = 0; g < 16; ++g) s += part[g][t];
    s *= (1.0f / (float)N_ROUTE);
    s_sh[t] = s;
    sq[t] = s * s;
  }
  __syncthreads();
  for (int off = 32; off > 0; off >>= 1) {
    if (t < off) sq[t] += sq[t + off];
    __syncthreads();
  }
  if (t < OUT_CH) {
    const float nsq = sq[0];
    const float scale = (nsq / (1.0f + nsq)) * rsqrtf(nsq);
    vbuf[(size_t)(c * BATCH + b) * OUT_CH + t] = scale * s_sh[t];
  }
}

// ---------------------------------------------------------------------------
// Kernel 3: one fused routing iteration per (c,b):
//   logits += priors . v_in ; probs = softmax_n ; v_out = squash(probs . priors)
// All per-lane priors traffic is b128; per-n dot reduced over 16-lane groups.
// ---------------------------------------------------------------------------
__global__ __launch_bounds__(256) void routing_iter_kernel(
    const float* __restrict__ priors, float* __restrict__ logits,
    const float* __restrict__ vin, float* __restrict__ vout) {
  const int b = blockIdx.x;
  const int c = blockIdx.y;
  const int t = threadIdx.x;
  const int lane = t & 31;
  const int wave = t >> 5;   // 0..7
  const int lr16 = lane & 15;
  const int lh = lane >> 4;
  const float* pr = priors + (size_t)(c * BATCH + b) * N_ROUTE * OUT_CH;
  float* lg = logits + (size_t)(c * BATCH + b) * N_ROUTE;

  __shared__ float vsh[OUT_CH];
  __shared__ float lsh[N_ROUTE];
  __shared__ float rbuf[256];

  if (t < OUT_CH) vsh[t] = vin[(size_t)(c * BATCH + b) * OUT_CH + t];
  __syncthreads();

  // delta[n] = sum_o priors[n][o]*v[o]; 2 n per wave pass (b128 per lane).
  const float4 vf = *(const float4*)&vsh[lr16 << 2];
  for (int nn = wave * 2; nn < N_ROUTE; nn += 16) {
    const int n1 = nn + lh;
    const float4 p = *(const float4*)(pr + (size_t)n1 * OUT_CH + (lr16 << 2));
    float d = p.x * vf.x + p.y * vf.y + p.z * vf.z + p.w * vf.w;
    d += __shfl_xor(d, 1);
    d += __shfl_xor(d, 2);
    d += __shfl_xor(d, 4);
    d += __shfl_xor(d, 8);
    if (lr16 == 0) {
      const float ln = lg[n1] + d;
      lsh[n1] = ln;
      lg[n1] = ln;  // persist for next iteration
    }
  }
  __syncthreads();

  // softmax over n
  float m = -3.402823466e38f;
  for (int nn = t; nn < N_ROUTE; nn += 256) m = fmaxf(m, lsh[nn]);
  rbuf[t] = m;
  __syncthreads();
  for (int off = 128; off > 0; off >>= 1) {
    if (t < off) rbuf[t] = fmaxf(rbuf[t], rbuf[t + off]);
    __syncthreads();
  }
  const float mx = rbuf[0];
  __syncthreads();

  float se = 0.0f;
  for (int nn = t; nn < N_ROUTE; nn += 256) {
    const float e = __expf(lsh[nn] - mx);
    lsh[nn] = e;
    se += e;
  }
  rbuf[t] = se;
  __syncthreads();
  for (int off = 128; off > 0; off >>= 1) {
    if (t < off) rbuf[t] += rbuf[t + off];
    __syncthreads();
  }
  const float inv = 1.0f / rbuf[0];

  // s[o] = sum_n probs[n]*priors[n][o]; b128 per lane, 16 n-groups.
  const int o4 = (t & 15) << 2;
  const int grp = t >> 4;
  float4 a4 = {0.f, 0.f, 0.f, 0.f};
  for (int nn = grp; nn < N_ROUTE; nn += 16) {
    const float pp = lsh[nn];
    const float4 p = *(const float4*)(pr + (size_t)nn * OUT_CH + o4);
    a4.x += pp * p.x; a4.y += pp * p.y; a4.z += pp * p.z; a4.w += pp * p.w;
  }
  a4.x *= inv; a4.y *= inv; a4.z *= inv; a4.w *= inv;
  __shared__ float part[16][OUT_CH];
  *(float4*)&part[grp][o4] = a4;
  __syncthreads();

  __shared__ float s_sh[OUT_CH];
  __shared__ float sq[OUT_CH];
  if (t < OUT_CH) {
    float s = 0.f;
#pragma unroll
    for (int g = 0; g < 16; ++g) s += part[g][t];
    s_sh[t] = s;
    sq[t] = s * s;
  }
  __syncthreads();
  for (int off = 32; off > 0; off >>= 1) {
    if (t < off) sq[t] += sq[t + off];
    __syncthreads();
  }
  if (t < OUT_CH) {
    const float nsq = sq[0];
    const float scale = (nsq / (1.0f + nsq)) * rsqrtf(nsq);
    vout[(size_t)(c * BATCH + b) * OUT_CH + t] = scale * s_sh[t];
  }
}

// ---------------------------------------------------------------------------
// Workspace (floats): priors [C,B,N,O] | v [C,B,O] | logits [C,B,N]
//                     | partials [C,NCHUNK,B,O]   (~345 MB total)
// ---------------------------------------------------------------------------
extern "C" void kernel_launch(void* const* d_in, const int* in_sizes, int n_in,
                              void* d_out, int out_size, void* d_ws,
                              size_t ws_size, hipStream_t stream) {
  (void)in_sizes; (void)n_in; (void)out_size; (void)ws_size;
  const float* x = (const float*)d_in[0];
  const float* rw = (const float*)d_in[1];
  float* priors = (float*)d_ws;
  const size_t P = (size_t)C_CAPS * BATCH * N_ROUTE * OUT_CH;
  float* vbuf = priors + P;
  float* lgts = vbuf + (size_t)C_CAPS * BATCH * OUT_CH;
  float* parts = lgts + (size_t)C_CAPS * BATCH * N_ROUTE;
  float* out = (float*)d_out;

  priors_gemm_kernel<<<dim3(NCHUNK, C_CAPS), 128, 0, stream>>>(x, rw, priors,
                                                               parts);
  routing_init_kernel<<<dim3(BATCH, C_CAPS), 256, 0, stream>>>(parts, vbuf,
                                                               lgts);
  routing_iter_kernel<<<dim3(BATCH, C_CAPS), 256, 0, stream>>>(priors, lgts,
                                                               vbuf, vbuf);
  routing_iter_kernel<<<dim3(BATCH, C_CAPS), 256, 0, stream>>>(priors, lgts,
                                                               vbuf, out);
}